// PPN_13408887898582
// MI455X (gfx1250) — compile-verified
//
#include <hip/hip_runtime.h>
#include <hip/hip_bf16.h>
#include <math.h>

// ---------------------------------------------------------------------------
// Problem constants (from reference): B=256, N=101, T=128, P=4, H=16
// ---------------------------------------------------------------------------
#define B_SZ 256
#define N_S  101
#define T_L  128
#define P_F  4
#define H_F  16
#define BN_ROWS (B_SZ * N_S)   // 25856 = 1616 * 16  (exact 16-row tiles)

typedef __attribute__((ext_vector_type(2))) float v2f;
typedef __attribute__((ext_vector_type(8))) float v8f;

// D = A(16x4) * B(4x16) + C(16x16), f32 WMMA (wave32).
__device__ __forceinline__ v8f wmma4(v2f a, v2f b, v8f c) {
  return __builtin_amdgcn_wmma_f32_16x16x4_f32(
      /*neg_a=*/false, a, /*neg_b=*/false, b,
      /*c_mod=*/(short)0, c, /*reuse_a=*/false, /*reuse_b=*/false);
}

// Branch-free activations on hardware transcendentals (v_exp_f32 is exp2).
#define LOG2E 1.44269504088896340736f
__device__ __forceinline__ float fast_sigmoid(float x) {
  float e = __builtin_amdgcn_exp2f(-x * LOG2E);          // exp(-x)
  return __builtin_amdgcn_rcpf(1.0f + e);
}
__device__ __forceinline__ float fast_tanh(float x) {
#if __has_builtin(__builtin_amdgcn_tanhf)
  return __builtin_amdgcn_tanhf(x);                      // V_TANH_F32
#else
  float e = __builtin_amdgcn_exp2f(2.0f * LOG2E * x);    // exp(2x)
  return 1.0f - 2.0f * __builtin_amdgcn_rcpf(e + 1.0f);
#endif
}

// ---------------------------------------------------------------------------
// Kernel 1: feature projection.  x[b,n,t,4] @ proj_w.T + proj_b
// Writes x_proj_f [BN, T, 4] (LSTM A-layout source) and
//        x_projT  [B, 4, N, T] (conv NCHW-style input).
// ---------------------------------------------------------------------------
__global__ void proj_kernel(const float* __restrict__ x,
                            const float* __restrict__ pw,
                            const float* __restrict__ pb,
                            float* __restrict__ xprojf,
                            float* __restrict__ xprojT) {
  size_t idx = (size_t)blockIdx.x * blockDim.x + threadIdx.x;  // over B*N*T
  size_t total = (size_t)BN_ROWS * T_L;
  if (idx >= total) return;
  size_t bn = idx / T_L;
  int t = (int)(idx % T_L);
  int b = (int)(bn / N_S);
  int n = (int)(bn % N_S);
  const float* xp = x + idx * P_F;
  float x0 = xp[0], x1 = xp[1], x2 = xp[2], x3 = xp[3];
#pragma unroll
  for (int o = 0; o < P_F; ++o) {
    float y = pb[o] + x0 * pw[o * 4 + 0] + x1 * pw[o * 4 + 1] +
              x2 * pw[o * 4 + 2] + x3 * pw[o * 4 + 3];
    xprojf[idx * P_F + o] = y;
    xprojT[(((size_t)b * P_F + o) * N_S + n) * T_L + t] = y;
  }
}

// ---------------------------------------------------------------------------
// Kernel 2: LSTM over T=128 steps.  One wave32 per 16-row tile (1616 tiles).
// Per step: g = bias + x_t @ Wih^T (K=4, 1 WMMA per gate tile)
//                + h   @ Whh^T (K=16, 4 WMMAs per gate tile)
// Gate tile ct directly equals gate (i,f,g,o), hidden index = lane&15.
// h is transposed C-layout -> A-layout through a padded LDS tile.
// x_{t+1} load is software-pipelined behind the 16 recurrent WMMAs.
// ---------------------------------------------------------------------------
__global__ void __launch_bounds__(32)
lstm_kernel(const float* __restrict__ xprojf,
            const float* __restrict__ w_ih,   // [64,4]
            const float* __restrict__ w_hh,   // [64,16]
            const float* __restrict__ b_ih,   // [64]
            const float* __restrict__ b_hh,   // [64]
            float* __restrict__ hlast) {      // [BN,16]
  __shared__ float hbuf[16 * 17];
  const int lane = threadIdx.x;
  const int half = lane >> 4;      // 0/1
  const int l15  = lane & 15;
  const int row0 = blockIdx.x * 16;

  // Preload B-layout weight tiles (lane L: N=l15, K = 2*half + {0,1})
  v2f Bx[4];
  v2f Bh[4][4];
  float bias[4];
#pragma unroll
  for (int ct = 0; ct < 4; ++ct) {
    int col = ct * 16 + l15;                 // gate column 0..63
    Bx[ct].x = w_ih[col * 4 + 2 * half];
    Bx[ct].y = w_ih[col * 4 + 2 * half + 1];
    bias[ct] = b_ih[col] + b_hh[col];
#pragma unroll
    for (int kk = 0; kk < 4; ++kk) {
      Bh[kk][ct].x = w_hh[col * 16 + kk * 4 + 2 * half];
      Bh[kk][ct].y = w_hh[col * 16 + kk * 4 + 2 * half + 1];
    }
  }

  // A-layout x_t source: lane L holds row (row0+l15), K pair at 2*half.
  const float* xrow = xprojf + ((size_t)(row0 + l15) * T_L) * P_F + 2 * half;

  float cst[8], hv[8];
#pragma unroll
  for (int i = 0; i < 8; ++i) { cst[i] = 0.f; hv[i] = 0.f; }

  v2f Ax;
  Ax.x = xrow[0];
  Ax.y = xrow[1];

  for (int t = 0; t < T_L; ++t) {
    // publish h (C-layout: VGPR i -> M=i+8*half, N=l15) into LDS [M][N]
#pragma unroll
    for (int i = 0; i < 8; ++i) hbuf[(i + 8 * half) * 17 + l15] = hv[i];
    __syncthreads();

    v8f acc[4];
#pragma unroll
    for (int ct = 0; ct < 4; ++ct) {
      v8f c0;
#pragma unroll
      for (int i = 0; i < 8; ++i) c0[i] = bias[ct];
      acc[ct] = wmma4(Ax, Bx[ct], c0);
    }

    // prefetch x_{t+1} (clamped) while the recurrent WMMAs run
    int tn = (t + 1 < T_L) ? (t + 1) : t;
    v2f Axn;
    Axn.x = xrow[(size_t)tn * P_F];
    Axn.y = xrow[(size_t)tn * P_F + 1];

#pragma unroll
    for (int kk = 0; kk < 4; ++kk) {
      v2f Ah;                                 // A-layout chunk of h
      Ah.x = hbuf[l15 * 17 + kk * 4 + 2 * half];
      Ah.y = hbuf[l15 * 17 + kk * 4 + 2 * half + 1];
#pragma unroll
      for (int ct = 0; ct < 4; ++ct) acc[ct] = wmma4(Ah, Bh[kk][ct], acc[ct]);
    }
    __syncthreads();  // hbuf reads done before next step's writes

#pragma unroll
    for (int i = 0; i < 8; ++i) {
      float ig = fast_sigmoid(acc[0][i]);
      float fg = fast_sigmoid(acc[1][i]);
      float gg = fast_tanh(acc[2][i]);
      float og = fast_sigmoid(acc[3][i]);
      cst[i] = fg * cst[i] + ig * gg;
      hv[i]  = og * fast_tanh(cst[i]);
    }
    Ax = Axn;
  }
#pragma unroll
  for (int i = 0; i < 8; ++i)
    hlast[(size_t)(row0 + i + 8 * half) * H_F + l15] = hv[i];
}

// ---------------------------------------------------------------------------
// Kernel 3: causal dilated 1x3 conv along T + ReLU, as implicit WMMA GEMM.
// out[b,o,n,t] = relu( sum_{c,tap} in[b,c,n, t-(2-tap)*DIL] * w[o,c,tap] )
// 4 waves/block, each wave one (b, n, 16-wide t tile).  M=o, N=t, K=3*Cin.
// All B loads are lane-contiguous along t.
// ---------------------------------------------------------------------------
template <int CIN, int COUT, int DIL>
__global__ void __launch_bounds__(128)
timeconv_kernel(const float* __restrict__ in,   // [B,CIN,N,T]
                const float* __restrict__ w,    // [COUT,CIN,1,3]
                float* __restrict__ out) {      // [B,COUT,N,T]
  const int lane = threadIdx.x;
  const int half = lane >> 4;
  const int l15  = lane & 15;
  const int t0 = (blockIdx.x * 4 + threadIdx.y) * 16;
  const int n  = blockIdx.y;
  const int b  = blockIdx.z;
  const float* inb = in + ((size_t)b * CIN) * ((size_t)N_S * T_L) + (size_t)n * T_L;

  v8f acc = {};
#pragma unroll
  for (int kk = 0; kk < CIN * 3; kk += 4) {
    const int k0 = kk + 2 * half;
    v2f a, bb;
    {
      int c = k0 / 3, tap = k0 % 3;
      a.x = (l15 < COUT) ? w[((size_t)l15 * CIN + c) * 3 + tap] : 0.f;
      int ts = t0 + l15 - (2 - tap) * DIL;
      bb.x = (ts >= 0) ? inb[(size_t)c * (N_S * T_L) + ts] : 0.f;
    }
    {
      int c = (k0 + 1) / 3, tap = (k0 + 1) % 3;
      a.y = (l15 < COUT) ? w[((size_t)l15 * CIN + c) * 3 + tap] : 0.f;
      int ts = t0 + l15 - (2 - tap) * DIL;
      bb.y = (ts >= 0) ? inb[(size_t)c * (N_S * T_L) + ts] : 0.f;
    }
    acc = wmma4(a, bb, acc);
  }
#pragma unroll
  for (int i = 0; i < 8; ++i) {
    int m = i + 8 * half;
    if (m < COUT) {
      float v = acc[i];
      out[(((size_t)b * COUT + m) * N_S + n) * T_L + t0 + l15] = v > 0.f ? v : 0.f;
    }
  }
}

// ---------------------------------------------------------------------------
// Kernel 4: cross-stock (11,1) conv along N + fused residual:
// out = relu( res + relu( conv11(in) ) ), res = 1x1 conv(resin) or resin.
// GEMM N-dim = time (lane-contiguous loads/stores); the 11 stock taps live
// in K.  4 waves/block, each wave one (b, n, 16-wide t tile).  K = 11*Cin.
// ---------------------------------------------------------------------------
template <int CIN, int COUT, int CRES, bool HAS_R>
__global__ void __launch_bounds__(128)
crossconv_kernel(const float* __restrict__ in,     // [B,CIN,N,T]
                 const float* __restrict__ w,      // [COUT,CIN,11,1]
                 const float* __restrict__ r,      // [COUT,CRES,1,1] or unused
                 const float* __restrict__ resin,  // [B,CRES,N,T]
                 float* __restrict__ out) {        // [B,COUT,N,T]
  const int lane = threadIdx.x;
  const int half = lane >> 4;
  const int l15  = lane & 15;
  const int t0 = (blockIdx.x * 4 + threadIdx.y) * 16;
  const int n  = blockIdx.y;
  const int b  = blockIdx.z;

  // main (11,1) conv accumulator
  v8f acc = {};
#pragma unroll 4
  for (int kk = 0; kk < CIN * 11; kk += 4) {
    const int k0 = kk + 2 * half;
    v2f a, bb;
    {
      int c = k0 / 11, dn = k0 % 11;
      a.x = (l15 < COUT) ? w[((size_t)l15 * CIN + c) * 11 + dn] : 0.f;
      int ns = n + dn - 5;
      bb.x = (ns >= 0 && ns < N_S)
                 ? in[(((size_t)b * CIN + c) * N_S + ns) * T_L + t0 + l15] : 0.f;
    }
    {
      int c = (k0 + 1) / 11, dn = (k0 + 1) % 11;
      a.y = (l15 < COUT) ? w[((size_t)l15 * CIN + c) * 11 + dn] : 0.f;
      int ns = n + dn - 5;
      bb.y = (ns >= 0 && ns < N_S)
                 ? in[(((size_t)b * CIN + c) * N_S + ns) * T_L + t0 + l15] : 0.f;
    }
    acc = wmma4(a, bb, acc);
  }

  // residual accumulator
  v8f racc = {};
  if (HAS_R) {
#pragma unroll
    for (int kk = 0; kk < CRES; kk += 4) {
      const int k0 = kk + 2 * half;
      v2f a, bb;
      a.x = (l15 < COUT) ? r[(size_t)l15 * CRES + k0] : 0.f;
      a.y = (l15 < COUT) ? r[(size_t)l15 * CRES + k0 + 1] : 0.f;
      bb.x = resin[(((size_t)b * CRES + k0) * N_S + n) * T_L + t0 + l15];
      bb.y = resin[(((size_t)b * CRES + k0 + 1) * N_S + n) * T_L + t0 + l15];
      racc = wmma4(a, bb, racc);
    }
  } else {
    // identity residual: CRES == COUT == 16, direct C-layout load
#pragma unroll
    for (int i = 0; i < 8; ++i) {
      int m = i + 8 * half;
      racc[i] = resin[(((size_t)b * CRES + m) * N_S + n) * T_L + t0 + l15];
    }
  }

#pragma unroll
  for (int i = 0; i < 8; ++i) {
    int m = i + 8 * half;
    if (m < COUT) {
      float h = acc[i];
      h = h > 0.f ? h : 0.f;            // inner relu
      float v = racc[i] + h;
      out[(((size_t)b * COUT + m) * N_S + n) * T_L + t0 + l15] = v > 0.f ? v : 0.f;
    }
  }
}

// ---------------------------------------------------------------------------
// Kernel 5: conv4 -- (1,T) conv collapsing time, + ReLU.  Pure WMMA GEMM:
// out[b,o,n] = relu( sum_{c,t} h3[b,c,n,t] * w4[o,c,0,t] ),  M=16, K=2048.
// 4 waves/block (over batch), each wave one (b, 16-wide n tile).
// ---------------------------------------------------------------------------
__global__ void __launch_bounds__(128)
conv4_kernel(const float* __restrict__ in,   // [B,16,N,T]
             const float* __restrict__ w4,   // [16,16,1,128]
             float* __restrict__ h4) {       // [B,16,N]
  const int lane = threadIdx.x;
  const int half = lane >> 4;
  const int l15  = lane & 15;
  const int n0 = blockIdx.x * 16;
  const int b  = blockIdx.y * 4 + threadIdx.y;
  const int ns = n0 + l15;
  const bool nok = (ns < N_S);

  v8f acc = {};
  for (int kk = 0; kk < 16 * T_L; kk += 4) {
    const int k0 = kk + 2 * half;
    v2f a, bb;
    {
      int c = k0 >> 7, tt = k0 & 127;
      a.x = w4[((size_t)l15 * 16 + c) * T_L + tt];
      bb.x = nok ? in[(((size_t)b * 16 + c) * N_S + ns) * T_L + tt] : 0.f;
    }
    {
      int c = (k0 + 1) >> 7, tt = (k0 + 1) & 127;
      a.y = w4[((size_t)l15 * 16 + c) * T_L + tt];
      bb.y = nok ? in[(((size_t)b * 16 + c) * N_S + ns) * T_L + tt] : 0.f;
    }
    acc = wmma4(a, bb, acc);
  }
#pragma unroll
  for (int i = 0; i < 8; ++i) {
    int m = i + 8 * half;
    if (nok) {
      float v = acc[i];
      h4[((size_t)b * 16 + m) * N_S + ns] = v > 0.f ? v : 0.f;
    }
  }
}

// ---------------------------------------------------------------------------
// Kernel 6: head -- concat(lstm reshape trick, h4) . out_w + out_b, softmax_N.
// One block (128 threads) per batch row b.
// Faithful torch reshape: z[b,n,h] = h_last[q % B, q / B, h], q = b*N + n.
// ---------------------------------------------------------------------------
__global__ void head_kernel(const float* __restrict__ hlast,  // [BN,16] as [B,N,16]
                            const float* __restrict__ h4,     // [B,16,N]
                            const float* __restrict__ out_w,  // [1,32]
                            const float* __restrict__ out_b,  // [1]
                            float* __restrict__ out) {        // [B,N]
  __shared__ float red[128];
  const int b = blockIdx.x;
  const int n = threadIdx.x;

  float logit = 0.f;
  if (n < N_S) {
    int q = b * N_S + n;
    int bsrc = q % B_SZ;
    int nsrc = q / B_SZ;
    const float* hp = hlast + ((size_t)bsrc * N_S + nsrc) * H_F;
    float l = out_b[0];
#pragma unroll
    for (int h = 0; h < H_F; ++h) l += hp[h] * out_w[h];
#pragma unroll
    for (int o = 0; o < 16; ++o)
      l += h4[((size_t)b * 16 + o) * N_S + n] * out_w[16 + o];
    logit = l;
  }

  // max-reduce
  red[n] = (n < N_S) ? logit : -INFINITY;
  __syncthreads();
  for (int s = 64; s > 0; s >>= 1) {
    if (n < s) red[n] = fmaxf(red[n], red[n + s]);
    __syncthreads();
  }
  float mx = red[0];
  __syncthreads();

  float e = (n < N_S) ? __builtin_amdgcn_exp2f((logit - mx) * LOG2E) : 0.f;
  red[n] = e;
  __syncthreads();
  for (int s = 64; s > 0; s >>= 1) {
    if (n < s) red[n] += red[n + s];
    __syncthreads();
  }
  float inv = 1.0f / red[0];
  if (n < N_S) out[(size_t)b * N_S + n] = e * inv;
}

// ---------------------------------------------------------------------------
// Host-side orchestration
// ---------------------------------------------------------------------------
extern "C" void kernel_launch(void* const* d_in, const int* in_sizes, int n_in,
                              void* d_out, int out_size, void* d_ws, size_t ws_size,
                              hipStream_t stream) {
  const float* x      = (const float*)d_in[0];
  const float* proj_w = (const float*)d_in[1];
  const float* proj_b = (const float*)d_in[2];
  const float* w_ih   = (const float*)d_in[3];
  const float* w_hh   = (const float*)d_in[4];
  const float* b_ih   = (const float*)d_in[5];
  const float* b_hh   = (const float*)d_in[6];
  const float* t1_d1  = (const float*)d_in[7];
  const float* t1_d2  = (const float*)d_in[8];
  const float* t1_c   = (const float*)d_in[9];
  const float* t1_r   = (const float*)d_in[10];
  const float* t2_d1  = (const float*)d_in[11];
  const float* t2_d2  = (const float*)d_in[12];
  const float* t2_c   = (const float*)d_in[13];
  const float* t2_r   = (const float*)d_in[14];
  const float* t3_d1  = (const float*)d_in[15];
  const float* t3_d2  = (const float*)d_in[16];
  const float* t3_c   = (const float*)d_in[17];
  const float* conv4w = (const float*)d_in[18];
  const float* out_w  = (const float*)d_in[19];
  const float* out_b  = (const float*)d_in[20];

  char* ws = (char*)d_ws;
  size_t off = 0;
  const size_t proj_elems = (size_t)BN_ROWS * T_L * P_F;       // 13.2M
  const size_t big_elems  = (size_t)B_SZ * 16 * N_S * T_L;     // 52.95M

  float* xprojf = (float*)(ws + off); off += proj_elems * sizeof(float);
  float* xprojT = (float*)(ws + off); off += proj_elems * sizeof(float);
  float* bufA   = (float*)(ws + off); off += big_elems * sizeof(float);
  float* bufB   = (float*)(ws + off); off += big_elems * sizeof(float);
  float* bufC   = (float*)(ws + off); off += big_elems * sizeof(float);
  float* hlast  = (float*)(ws + off); off += (size_t)BN_ROWS * H_F * sizeof(float);
  float* h4     = (float*)(ws + off); off += (size_t)B_SZ * 16 * N_S * sizeof(float);
  (void)ws_size; (void)in_sizes; (void)n_in; (void)out_size;

  // 1) projection
  {
    size_t total = (size_t)BN_ROWS * T_L;
    int blocks = (int)((total + 255) / 256);
    proj_kernel<<<blocks, 256, 0, stream>>>(x, proj_w, proj_b, xprojf, xprojT);
  }

  // 2) LSTM (1616 tiles of 16 rows, one wave32 each)
  lstm_kernel<<<BN_ROWS / 16, 32, 0, stream>>>(xprojf, w_ih, w_hh, b_ih, b_hh, hlast);

  dim3 blk(32, 4);
  dim3 tg(T_L / 64, N_S, B_SZ);     // time/cross conv grid: 2 x 101 x 256

  // 3) TCCB block 1: 4 -> 8 channels, dil=1
  timeconv_kernel<4, 8, 1><<<tg, blk, 0, stream>>>(xprojT, t1_d1, bufA);
  timeconv_kernel<8, 8, 1><<<tg, blk, 0, stream>>>(bufA, t1_d2, bufB);
  crossconv_kernel<8, 8, 4, true><<<tg, blk, 0, stream>>>(bufB, t1_c, t1_r, xprojT, bufC);

  // 4) TCCB block 2: 8 -> 16 channels, dil=2
  timeconv_kernel<8, 16, 2><<<tg, blk, 0, stream>>>(bufC, t2_d1, bufA);
  timeconv_kernel<16, 16, 2><<<tg, blk, 0, stream>>>(bufA, t2_d2, bufB);
  crossconv_kernel<16, 16, 8, true><<<tg, blk, 0, stream>>>(bufB, t2_c, t2_r, bufC, bufA);

  // 5) TCCB block 3: 16 -> 16 channels, dil=4, identity residual
  timeconv_kernel<16, 16, 4><<<tg, blk, 0, stream>>>(bufA, t3_d1, bufB);
  timeconv_kernel<16, 16, 4><<<tg, blk, 0, stream>>>(bufB, t3_d2, bufC);
  crossconv_kernel<16, 16, 16, false><<<tg, blk, 0, stream>>>(bufC, t3_c, nullptr, bufA, bufB);

  // 6) conv4: collapse T (K=2048 GEMM), 4 waves/block over batch
  {
    dim3 g((N_S + 15) / 16, B_SZ / 4);
    conv4_kernel<<<g, blk, 0, stream>>>(bufB, conv4w, h4);
  }

  // 7) head: linear + softmax over stocks
  head_kernel<<<B_SZ, 128, 0, stream>>>(hlast, h4, out_w, out_b, (float*)d_out);
}